// RNN_395136991830
// MI455X (gfx1250) — compile-verified
//
#include <hip/hip_runtime.h>
#include <hip/hip_bf16.h>
#include <math.h>

#define SEQ   512
#define BATCH 1024
#define NFEAT 128
#define HID   20
#define NCLS  10

typedef __attribute__((ext_vector_type(2))) float v2f;
typedef __attribute__((ext_vector_type(8))) float v8f;

// D = A(16x4 f32) * B(4x16 f32) + C(16x16 f32)
static __device__ __forceinline__ v8f wmma_f32(v2f a, v2f b, v8f c) {
    return __builtin_amdgcn_wmma_f32_16x16x4_f32(false, a, false, b, (short)0, c,
                                                 false, false);
}

// ---------------------------------------------------------------------------
// Kernel 1: xp[t*B+b][h] = sum_f x[t][b][f]*W_ih[h][f] + b_ih[h] + b_hh[h]
// M = SEQ*BATCH rows tiled 16 per wave, N = 20 as two 16-wide WMMA tiles,
// K = 128 as 32 steps of K=4 fp32 WMMA. W_ih staged in LDS (padded to 32
// rows of zeros, stride 132 floats for bank-conflict-free b64 reads).
// ---------------------------------------------------------------------------
#define WSTRIDE 132

__global__ __launch_bounds__(256) void xp_gemm_kernel(
    const float* __restrict__ x, const float* __restrict__ W_ih,
    const float* __restrict__ b_ih, const float* __restrict__ b_hh,
    float* __restrict__ xp)
{
    __shared__ float Wl[32 * WSTRIDE];
    __shared__ float bl[32];
    const int tid = threadIdx.x;

    for (int i = tid; i < 32 * NFEAT; i += 256) {
        int r = i >> 7, c = i & (NFEAT - 1);
        Wl[r * WSTRIDE + c] = (r < HID) ? W_ih[r * NFEAT + c] : 0.0f;
    }
    if (tid < 32) bl[tid] = (tid < HID) ? (b_ih[tid] + b_hh[tid]) : 0.0f;
    __syncthreads();

    const int lane  = tid & 31;
    const int wave  = tid >> 5;
    const int m0    = (blockIdx.x * 8 + wave) * 16;   // 16-row M tile
    const int mloc  = lane & 15;
    const int khalf = (lane >> 4) * 2;                // 0 or 2 (A/B frag K split)
    const int n0    = mloc;                           // N tile 0: cols 0..15
    const int n1    = 16 + mloc;                      // N tile 1: cols 16..31 (20..31 dead)

    const float* arow = x + (size_t)(m0 + mloc) * NFEAT + khalf;

    v8f acc0 = {};
    v8f acc1 = {};
    for (int k0 = 0; k0 < NFEAT; k0 += 4) {
        v2f a  = *(const v2f*)(arow + k0);
        v2f b0 = *(const v2f*)(&Wl[n0 * WSTRIDE + k0 + khalf]);
        v2f b1 = *(const v2f*)(&Wl[n1 * WSTRIDE + k0 + khalf]);
        acc0 = wmma_f32(a, b0, acc0);
        acc1 = wmma_f32(a, b1, acc1);
    }

    // C/D layout: VGPR i holds row (i + 8*(lane>>4)), col lane&15
    const int rbase = m0 + ((lane >> 4) << 3);
    const float bias0 = bl[n0];
    const float bias1 = bl[(n1 < HID) ? n1 : 0];
    #pragma unroll
    for (int i = 0; i < 8; ++i) {
        size_t row = (size_t)(rbase + i);
        xp[row * HID + n0] = acc0[i] + bias0;
        if (n1 < HID) xp[row * HID + n1] = acc1[i] + bias1;
    }
}

// ---------------------------------------------------------------------------
// Kernel 2: sequential scan. Each single-wave block owns 16 batch rows for
// all 512 timesteps (recurrence independent across batch -> no global sync).
// h_t = tanh(xp_t + h_{t-1} @ W_hh^T) via 5x2 fp32 WMMAs; W_hh fragments
// live in registers; h transits LDS for the D->A layout transpose.
// xp_{t+1} is prefetched into registers one step ahead. FC folded at the end.
// ---------------------------------------------------------------------------
#define HSTRIDE 26   // floats; conflict-free & 8B-aligned for v2f reads

__global__ __launch_bounds__(32) void rnn_scan_kernel(
    const float* __restrict__ xp, const float* __restrict__ W_hh,
    const float* __restrict__ W_fc, const float* __restrict__ b_fc,
    float* __restrict__ out)
{
    __shared__ float h[16 * HSTRIDE];
    const int lane  = threadIdx.x;
    const int b0    = blockIdx.x * 16;
    const int mloc  = lane & 15;
    const int khalf = (lane >> 4) * 2;
    const int n0    = mloc;
    const int n1    = 16 + mloc;
    const int c1    = (n1 < HID) ? n1 : (HID - 1);   // clamped (dead lanes)
    const int rloc  = (lane >> 4) << 3;              // 0 or 8

    // Preload B fragments of W_hh: B[k][n] = W_hh[n][k], K=20 -> 5 frags x 2 tiles
    v2f wb0[5], wb1[5];
    #pragma unroll
    for (int kt = 0; kt < 5; ++kt) {
        int kk = kt * 4 + khalf;
        wb0[kt].x = W_hh[n0 * HID + kk];
        wb0[kt].y = W_hh[n0 * HID + kk + 1];
        if (n1 < HID) {
            wb1[kt].x = W_hh[n1 * HID + kk];
            wb1[kt].y = W_hh[n1 * HID + kk + 1];
        } else {
            wb1[kt].x = 0.0f; wb1[kt].y = 0.0f;
        }
    }

    for (int i = lane; i < 16 * HSTRIDE; i += 32) h[i] = 0.0f;  // h_{-1} = 0
    __syncthreads();

    // Prefetch xp for t = 0 (C-layout registers)
    float xa0[8], xa1[8];
    {
        const float* p = xp + (size_t)(b0 + rloc) * HID;
        #pragma unroll
        for (int i = 0; i < 8; ++i) {
            xa0[i] = p[(size_t)i * HID + n0];
            xa1[i] = p[(size_t)i * HID + c1];
        }
    }

    for (int t = 0; t < SEQ; ++t) {
        v8f acc0, acc1;
        #pragma unroll
        for (int i = 0; i < 8; ++i) { acc0[i] = xa0[i]; acc1[i] = xa1[i]; }

        // Prefetch next timestep's xp while this step computes
        if (t + 1 < SEQ) {
            const float* p = xp + ((size_t)(t + 1) * BATCH + b0 + rloc) * HID;
            #pragma unroll
            for (int i = 0; i < 8; ++i) {
                xa0[i] = p[(size_t)i * HID + n0];
                xa1[i] = p[(size_t)i * HID + c1];
            }
        }

        // h_{t-1} @ W_hh^T : A frags from LDS (M=mloc, K=kt*4+khalf)
        #pragma unroll
        for (int kt = 0; kt < 5; ++kt) {
            int kk = kt * 4 + khalf;
            v2f a = *(const v2f*)(&h[mloc * HSTRIDE + kk]);
            acc0 = wmma_f32(a, wb0[kt], acc0);
            acc1 = wmma_f32(a, wb1[kt], acc1);
        }

        // tanh and write h_t back to LDS (D layout -> [row][col])
        #pragma unroll
        for (int i = 0; i < 8; ++i) {
            h[(rloc + i) * HSTRIDE + n0] = tanhf(acc0[i]);
            if (n1 < HID) h[(rloc + i) * HSTRIDE + n1] = tanhf(acc1[i]);
        }
        __syncthreads();   // single-wave WG: lowers to a dscnt wait
    }

    // FC on h_last: 16 batches x 10 classes = 160 outputs, 5 per lane
    #pragma unroll
    for (int j = 0; j < 5; ++j) {
        int idx = j * 32 + lane;            // 0..159
        int bl  = idx / NCLS;
        int c   = idx - bl * NCLS;
        float s = b_fc[c];
        #pragma unroll
        for (int k = 0; k < HID; ++k)
            s += h[bl * HSTRIDE + k] * W_fc[c * HID + k];
        out[(size_t)(b0 + bl) * NCLS + c] = s;
    }
}

// ---------------------------------------------------------------------------
extern "C" void kernel_launch(void* const* d_in, const int* in_sizes, int n_in,
                              void* d_out, int out_size, void* d_ws, size_t ws_size,
                              hipStream_t stream)
{
    const float* x    = (const float*)d_in[0];
    const float* W_ih = (const float*)d_in[1];
    const float* W_hh = (const float*)d_in[2];
    const float* b_ih = (const float*)d_in[3];
    const float* b_hh = (const float*)d_in[4];
    const float* W_fc = (const float*)d_in[5];
    const float* b_fc = (const float*)d_in[6];
    float* out = (float*)d_out;
    float* xp  = (float*)d_ws;   // SEQ*BATCH*HID*4 = 40 MB scratch

    // 4096 blocks x 8 waves x 16 rows = 524288 = SEQ*BATCH
    hipLaunchKernelGGL(xp_gemm_kernel, dim3((SEQ * BATCH) / (8 * 16)), dim3(256),
                       0, stream, x, W_ih, b_ih, b_hh, xp);
    // 64 single-wave blocks, each owns 16 batch rows for all timesteps
    hipLaunchKernelGGL(rnn_scan_kernel, dim3(BATCH / 16), dim3(32),
                       0, stream, xp, W_hh, W_fc, b_fc, out);
}